// Net_Jentzen_1network_42614665511485
// MI455X (gfx1250) — compile-verified
//
#include <hip/hip_runtime.h>

#define Bsz   2048
#define DIMC  256
#define HC    356
#define HPC   384      // H padded to multiple of 32
#define KTXC  260      // (DIM+1)=257 padded to multiple of 4 (even stride)
#define TC    50
#define KAPPA_C 1.0f
#define SIGMA_C 0.3f
#define BN_EPS_C 1e-5f
#define RCHUNK 16
#define KMAX  384      // max K across all GEMMs (even)
#define PBK   64       // floats per K-pair row in LDS panel (32 cols x float2)

typedef __attribute__((ext_vector_type(2))) float v2f;
typedef __attribute__((ext_vector_type(8))) float v8f;

// ---------------- fp32 WMMA GEMM:  C[B x N] = A[B x K] @ W[K x N] + bias[N]
// K multiple of 4 (<=384), N multiple of 32.
// Block = 8 waves = 128 rows x 32 cols. Weight panel (K x 32) staged in LDS
// in pair-interleaved "fragment-ready" layout: P[k/2][n] = (W[k][n], W[k+1][n]).
// Inner loop: 1 global b64 (A) + 1 ds_load_2addr_b64 (B fragments) + 2 WMMAs.
__global__ void __launch_bounds__(256)
gemm_wmma(const float* __restrict__ A, const float* __restrict__ W,
          const float* __restrict__ bias, float* __restrict__ C,
          int N, int K) {
  __shared__ float Wp[(KMAX / 2) * PBK];   // 192*64*4 = 48KB

  const int tid  = threadIdx.x;
  const int col0 = blockIdx.x << 5;

  // ---- cooperative interleaved stage of W[K x 32] panel into LDS
  {
    const int rpin = tid >> 4;           // 0..15 pair-row within pass
    const int c2   = tid & 15;           // column pair 0..15 -> cols 2c2,2c2+1
    const int K2   = K >> 1;
    for (int rp = rpin; rp < K2; rp += 16) {
      const float2 w0 = *(const float2*)(W + (size_t)(2 * rp)     * N + col0 + 2 * c2);
      const float2 w1 = *(const float2*)(W + (size_t)(2 * rp + 1) * N + col0 + 2 * c2);
      float4 p; p.x = w0.x; p.y = w1.x; p.z = w0.y; p.w = w1.y;
      *(float4*)(Wp + rp * PBK + 4 * c2) = p;
    }
  }
  __syncthreads();

  const int wave = tid >> 5;
  const int lane = tid & 31;
  const int lh = lane >> 4;            // lane-half selects K+0 / K+2
  const int lm = lane & 15;
  const int row0 = blockIdx.y * 128 + (wave << 4);

  const float* __restrict__ arow = A + (size_t)(row0 + lm) * K + 2 * lh;
  const int nA = col0 + lm;
  const int nB = nA + 16;
  const float* __restrict__ bbase = Wp + lh * PBK + 2 * lm;

  v8f acc0 = {0.f,0.f,0.f,0.f,0.f,0.f,0.f,0.f};
  v8f acc1 = {0.f,0.f,0.f,0.f,0.f,0.f,0.f,0.f};
  for (int k = 0; k < K; k += 4) {
    v2f a = *(const v2f*)(arow + k);          // A: M=lane&15, K = k+2lh, k+2lh+1
    const float* bp = bbase + (k >> 1) * PBK; // pair-row k/2 (+lh)
    v2f b0 = *(const v2f*)(bp);               // {W[kr][nA], W[kr+1][nA]}
    v2f b1 = *(const v2f*)(bp + 32);          // {W[kr][nB], W[kr+1][nB]}
    acc0 = __builtin_amdgcn_wmma_f32_16x16x4_f32(false, a, false, b0, (short)0, acc0, false, false);
    acc1 = __builtin_amdgcn_wmma_f32_16x16x4_f32(false, a, false, b1, (short)0, acc1, false, false);
  }
  const float bias0 = bias[nA];
  const float bias1 = bias[nB];
#pragma unroll
  for (int i = 0; i < 8; ++i) {
    const int r = row0 + i + 8 * lh;          // D layout: VGPR i -> M = i / 8+i
    C[(size_t)r * N + nA] = acc0[i] + bias0;
    C[(size_t)r * N + nB] = acc1[i] + bias1;
  }
}

// ---------------- batch-norm statistics (two stage, over 2048 rows per column)
__global__ void colstats_partial(const float* __restrict__ X, float* __restrict__ ps,
                                 float* __restrict__ ps2, int N) {
  int j = blockIdx.x * blockDim.x + threadIdx.x;
  int chunk = blockIdx.y;
  if (j >= N) return;
  int r0 = chunk * (Bsz / RCHUNK);
  float s = 0.f, s2 = 0.f;
  for (int r = 0; r < Bsz / RCHUNK; ++r) {
    float v = X[(size_t)(r0 + r) * N + j];
    s += v; s2 += v * v;
  }
  ps[chunk * N + j]  = s;
  ps2[chunk * N + j] = s2;
}

__global__ void colstats_final(const float* __restrict__ ps, const float* __restrict__ ps2,
                               float* __restrict__ mu, float* __restrict__ rs, int N) {
  int j = blockIdx.x * blockDim.x + threadIdx.x;
  if (j >= N) return;
  float s = 0.f, s2 = 0.f;
  for (int c = 0; c < RCHUNK; ++c) { s += ps[c * N + j]; s2 += ps2[c * N + j]; }
  float m   = s * (1.0f / Bsz);
  float var = s2 * (1.0f / Bsz) - m * m;
  mu[j] = m;
  rs[j] = rsqrtf(fmaxf(var, 0.f) + BN_EPS_C);
}

__global__ void bn_relu(float* __restrict__ X, const float* __restrict__ mu,
                        const float* __restrict__ rs, const float* __restrict__ g,
                        const float* __restrict__ be, int N) {
  int idx = blockIdx.x * blockDim.x + threadIdx.x;
  if (idx >= Bsz * N) return;
  int j = idx % N;
  float y = (X[idx] - mu[j]) * rs[j] * g[j] + be[j];
  X[idx] = fmaxf(y, 0.f);
}

// ---------------- 356->1 projection for v0 (wave per row)
__global__ void v0_dot(const float* __restrict__ Xact, const float* __restrict__ w6,
                       const float* __restrict__ b6, float* __restrict__ v) {
  int row  = (blockIdx.x * blockDim.x + threadIdx.x) >> 5;
  int lane = threadIdx.x & 31;
  if (row >= Bsz) return;
  float s = 0.f;
  for (int k = lane; k < HPC; k += 32) s += Xact[(size_t)row * HPC + k] * w6[k];
  for (int off = 16; off; off >>= 1) s += __shfl_down(s, off);
  if (lane == 0) v[row] = s + b6[0];
}

// ---------------- build tx = [t_i, x_c, zero-pad]  (B x 260)
__global__ void build_tx(const float* __restrict__ xc, const float* __restrict__ tg,
                         int step, float* __restrict__ TX) {
  int idx = blockIdx.x * blockDim.x + threadIdx.x;
  if (idx >= Bsz * KTXC) return;
  int r = idx / KTXC, c = idx - r * KTXC;
  float val;
  if (c == 0)          val = tg[step];
  else if (c <= DIMC)  val = xc[(size_t)r * DIMC + (c - 1)];
  else                 val = 0.f;
  TX[idx] = val;
}

// ---------------- SDE state update (wave per row)
__global__ void step_update(const float* __restrict__ xc, const float* __restrict__ alpha,
                            const float* __restrict__ grad, const float* __restrict__ law_i,
                            const float* __restrict__ xi_i, const float* __restrict__ tg,
                            int step, float* __restrict__ v, float* __restrict__ x_new) {
  int row  = (blockIdx.x * blockDim.x + threadIdx.x) >> 5;
  int lane = threadIdx.x & 31;
  if (row >= Bsz) return;
  float hi    = tg[step + 1] - tg[step];
  float scale = SIGMA_C * sqrtf(hi);
  float f1 = 0.f, f2 = 0.f, dot = 0.f;
  for (int c = lane; c < DIMC; c += 32) {
    size_t o = (size_t)row * DIMC + c;
    float x  = xc[o];
    float d  = x - law_i[c];
    f1 += d * d;
    float al = alpha[o];
    f2 += al * al;
    float nw = scale * xi_i[o];
    dot += grad[o] * nw;
    x_new[o] = x + al * hi + nw;
  }
  for (int off = 16; off; off >>= 1) {
    f1  += __shfl_down(f1, off);
    f2  += __shfl_down(f2, off);
    dot += __shfl_down(dot, off);
  }
  if (lane == 0) v[row] += -(0.5f * KAPPA_C * f1 + 0.5f * f2) * hi + dot;
}

// ---------------- zero-padding / copy helpers
__global__ void pad2d(const float* __restrict__ src, float* __restrict__ dst,
                      int K, int N, int Kp, int Np) {
  int idx = blockIdx.x * blockDim.x + threadIdx.x;
  if (idx >= Kp * Np) return;
  int r = idx / Np, c = idx - r * Np;
  dst[idx] = (r < K && c < N) ? src[r * N + c] : 0.f;
}
__global__ void pad1d(const float* __restrict__ src, float* __restrict__ dst, int N, int Np) {
  int j = blockIdx.x * blockDim.x + threadIdx.x;
  if (j >= Np) return;
  dst[j] = (j < N) ? src[j] : 0.f;
}
__global__ void copyf(const float* __restrict__ src, float* __restrict__ dst, int n) {
  int i = blockIdx.x * blockDim.x + threadIdx.x;
  if (i < n) dst[i] = src[i];
}

extern "C" void kernel_launch(void* const* d_in, const int* in_sizes, int n_in,
                              void* d_out, int out_size, void* d_ws, size_t ws_size,
                              hipStream_t stream) {
  (void)in_sizes; (void)n_in; (void)out_size; (void)ws_size;
  const float* x   = (const float*)d_in[0];
  const float* xi  = (const float*)d_in[1];
  const float* law = (const float*)d_in[2];
  const float* tg  = (const float*)d_in[3];
  const float* Wa  = (const float*)d_in[4];
  const float* ba  = (const float*)d_in[5];
  const float* W1  = (const float*)d_in[6];
  const float* b1  = (const float*)d_in[7];
  const float* g1  = (const float*)d_in[8];
  const float* be1 = (const float*)d_in[9];
  const float* W2  = (const float*)d_in[10];
  const float* b2  = (const float*)d_in[11];
  const float* g2  = (const float*)d_in[12];
  const float* be2 = (const float*)d_in[13];
  const float* W3  = (const float*)d_in[14];
  const float* b3  = (const float*)d_in[15];
  const float* W4  = (const float*)d_in[16];
  const float* b4  = (const float*)d_in[17];
  const float* g4  = (const float*)d_in[18];
  const float* be4 = (const float*)d_in[19];
  const float* W5  = (const float*)d_in[20];
  const float* b5  = (const float*)d_in[21];
  const float* g5  = (const float*)d_in[22];
  const float* be5 = (const float*)d_in[23];
  const float* W6  = (const float*)d_in[24];
  const float* b6  = (const float*)d_in[25];

  float* out    = (float*)d_out;
  float* v_out  = out;                               // (B,1)
  float* xT_out = out + Bsz;                         // (B,DIM)
  float* path   = xT_out + (size_t)Bsz * DIMC;       // (T+1,B,DIM)

  float* ws = (float*)d_ws;
  size_t off = 0;
  auto alloc = [&](size_t n) { float* p = ws + off; off += n; return p; };

  float* WaP = alloc((size_t)KTXC * DIMC);
  float* W1P = alloc((size_t)KTXC * HPC);
  float* W2P = alloc((size_t)HPC  * HPC);
  float* W3P = alloc((size_t)HPC  * DIMC);
  float* W4P = alloc((size_t)DIMC * HPC);
  float* W5P = alloc((size_t)HPC  * HPC);
  float* W6P = alloc(HPC);
  float* b1P = alloc(HPC);  float* b2P = alloc(HPC);
  float* b4P = alloc(HPC);  float* b5P = alloc(HPC);
  float* g1P = alloc(HPC);  float* be1P = alloc(HPC);
  float* g2P = alloc(HPC);  float* be2P = alloc(HPC);
  float* g4P = alloc(HPC);  float* be4P = alloc(HPC);
  float* g5P = alloc(HPC);  float* be5P = alloc(HPC);
  float* TX    = alloc((size_t)Bsz * KTXC);
  float* ACT1  = alloc((size_t)Bsz * HPC);
  float* ACT2  = alloc((size_t)Bsz * HPC);
  float* ALPHA = alloc((size_t)Bsz * DIMC);
  float* GRAD  = alloc((size_t)Bsz * DIMC);
  float* MU  = alloc(HPC);  float* RS  = alloc(HPC);
  float* PS  = alloc((size_t)RCHUNK * HPC);
  float* PS2 = alloc((size_t)RCHUNK * HPC);

  auto p2 = [&](const float* s, float* d, int K, int N, int Kp, int Np) {
    int tot = Kp * Np;
    pad2d<<<(tot + 255) / 256, 256, 0, stream>>>(s, d, K, N, Kp, Np);
  };
  auto p1 = [&](const float* s, float* d, int N, int Np) {
    pad1d<<<(Np + 255) / 256, 256, 0, stream>>>(s, d, N, Np);
  };
  auto gemm = [&](const float* A, const float* Wp, const float* bp, float* C, int N, int K) {
    dim3 grid(N / 32, Bsz / 128);
    gemm_wmma<<<grid, 256, 0, stream>>>(A, Wp, bp, C, N, K);
  };
  auto bn = [&](float* X, const float* g, const float* be, int N) {
    dim3 gp((N + 255) / 256, RCHUNK);
    colstats_partial<<<gp, 256, 0, stream>>>(X, PS, PS2, N);
    colstats_final<<<(N + 255) / 256, 256, 0, stream>>>(PS, PS2, MU, RS, N);
    int tot = Bsz * N;
    bn_relu<<<(tot + 255) / 256, 256, 0, stream>>>(X, MU, RS, g, be, N);
  };

  // ---- pad weights / vectors (done every call; deterministic)
  p2(Wa, WaP, DIMC + 1, DIMC, KTXC, DIMC);
  p2(W1, W1P, DIMC + 1, HC,   KTXC, HPC);
  p2(W2, W2P, HC, HC,   HPC, HPC);
  p2(W3, W3P, HC, DIMC, HPC, DIMC);
  p2(W4, W4P, DIMC, HC, DIMC, HPC);
  p2(W5, W5P, HC, HC,   HPC, HPC);
  p1(W6, W6P, HC, HPC);
  p1(b1, b1P, HC, HPC);  p1(b2, b2P, HC, HPC);
  p1(b4, b4P, HC, HPC);  p1(b5, b5P, HC, HPC);
  p1(g1, g1P, HC, HPC);  p1(be1, be1P, HC, HPC);
  p1(g2, g2P, HC, HPC);  p1(be2, be2P, HC, HPC);
  p1(g4, g4P, HC, HPC);  p1(be4, be4P, HC, HPC);
  p1(g5, g5P, HC, HPC);  p1(be5, be5P, HC, HPC);

  // ---- initial value network: v0
  gemm(x, W4P, b4P, ACT1, HPC, DIMC);
  bn(ACT1, g4P, be4P, HPC);
  gemm(ACT1, W5P, b5P, ACT2, HPC, HPC);
  bn(ACT2, g5P, be5P, HPC);
  v0_dot<<<(Bsz * 32) / 256, 256, 0, stream>>>(ACT2, W6P, b6, v_out);

  // ---- path[0] = x
  copyf<<<((Bsz * DIMC) + 255) / 256, 256, 0, stream>>>(x, path, Bsz * DIMC);

  // ---- sequential scan over T steps; x state lives in the path output
  for (int i = 0; i < TC; ++i) {
    const float* xc = path + (size_t)i * Bsz * DIMC;
    float*       xn = path + (size_t)(i + 1) * Bsz * DIMC;
    build_tx<<<((Bsz * KTXC) + 255) / 256, 256, 0, stream>>>(xc, tg, i, TX);
    gemm(TX, WaP, ba, ALPHA, DIMC, KTXC);
    gemm(TX, W1P, b1P, ACT1, HPC, KTXC);
    bn(ACT1, g1P, be1P, HPC);
    gemm(ACT1, W2P, b2P, ACT2, HPC, HPC);
    bn(ACT2, g2P, be2P, HPC);
    gemm(ACT2, W3P, b3, GRAD, DIMC, HPC);
    step_update<<<(Bsz * 32) / 256, 256, 0, stream>>>(
        xc, ALPHA, GRAD, law + (size_t)i * DIMC, xi + (size_t)i * Bsz * DIMC,
        tg, i, v_out, xn);
  }

  // ---- x_T = path[T]
  copyf<<<((Bsz * DIMC) + 255) / 256, 256, 0, stream>>>(
      path + (size_t)TC * Bsz * DIMC, xT_out, Bsz * DIMC);
}